// MultiTimeAttention_21242908246365
// MI455X (gfx1250) — compile-verified
//
#include <hip/hip_runtime.h>
#include <math.h>

// ---- problem constants ----
#define NB   8
#define NLQ  128
#define NLK  256
#define ND   32
#define NET  32
#define NH_  8     // heads
#define NNH  64    // hidden
#define LN_EPS 1e-5f

typedef __attribute__((ext_vector_type(2))) float v2f;
typedef __attribute__((ext_vector_type(8))) float v8f;

// D = A(16x4) * B(4x16) + C, fp32 WMMA (CDNA5 V_WMMA_F32_16X16X4_F32)
__device__ __forceinline__ v8f wmma_f32(v2f a, v2f b, v8f c) {
  return __builtin_amdgcn_wmma_f32_16x16x4_f32(false, a, false, b, (short)0, c,
                                               false, false);
}

// =====================================================================
// Kernel 1: per-(b,k) row:  value -> w1 -> LN -> relu -> w2  (dvalue),
// then emit masked matrices MV = mask*value, Mf = mask, MD = mask*dvalue.
// One wave32 per row; NH=64 hidden -> 2 per lane; all-lane data via shfl.
// =====================================================================
__global__ __launch_bounds__(32) void prep_value_kernel(
    const float* __restrict__ value, const int* __restrict__ emb_mask,
    const float* __restrict__ w1, const float* __restrict__ b1,
    const float* __restrict__ ln_g, const float* __restrict__ ln_b,
    const float* __restrict__ w2, const float* __restrict__ b2,
    float* __restrict__ MV, float* __restrict__ Mf, float* __restrict__ MD)
{
  const int row  = blockIdx.x;          // b*LK + k
  const int lane = threadIdx.x;
  const float myv = value[(size_t)row * ND + lane];

  float h0 = b1[lane], h1 = b1[lane + 32];
  #pragma unroll 4
  for (int d = 0; d < ND; ++d) {
    const float vd = __shfl(myv, d, 32);
    h0 = fmaf(vd, w1[d * NNH + lane],      h0);
    h1 = fmaf(vd, w1[d * NNH + lane + 32], h1);
  }
  // LayerNorm over 64
  float s = h0 + h1;
  for (int o = 16; o > 0; o >>= 1) s += __shfl_xor(s, o, 32);
  const float mu = s * (1.0f / 64.0f);
  const float d0 = h0 - mu, d1 = h1 - mu;
  float vs = d0 * d0 + d1 * d1;
  for (int o = 16; o > 0; o >>= 1) vs += __shfl_xor(vs, o, 32);
  const float inv = rsqrtf(vs * (1.0f / 64.0f) + LN_EPS);
  float r0 = fmaxf(fmaf(d0 * inv, ln_g[lane],      ln_b[lane]),      0.0f);
  float r1 = fmaxf(fmaf(d1 * inv, ln_g[lane + 32], ln_b[lane + 32]), 0.0f);

  float dv = b2[lane];                  // dvalue[row][d=lane]
  #pragma unroll 4
  for (int j = 0; j < 32; ++j) {
    dv = fmaf(__shfl(r0, j, 32), w2[j * ND + lane],        dv);
    dv = fmaf(__shfl(r1, j, 32), w2[(j + 32) * ND + lane], dv);
  }

  const float fm = emb_mask[(size_t)row * ND + lane] ? 1.0f : 0.0f;
  const size_t o = (size_t)row * ND + lane;
  MV[o] = fm * myv;
  Mf[o] = fm;
  MD[o] = fm * dv;
}

// =====================================================================
// Kernel 2: time embedding + 32x32 projection. One wave per row.
// te = [t*w_time+b_time, sin(t*w_per+b_per)];  out = te @ W + bias
// =====================================================================
__global__ __launch_bounds__(32) void time_emb_kernel(
    const float* __restrict__ tt,
    const float* __restrict__ w_time, const float* __restrict__ b_time,
    const float* __restrict__ w_per,  const float* __restrict__ b_per,
    const float* __restrict__ W,      const float* __restrict__ Bias,
    float* __restrict__ out)
{
  const int row = blockIdx.x, lane = threadIdx.x;
  const float t = tt[row];
  const float te = (lane == 0)
      ? fmaf(t, w_time[0], b_time[0])
      : sinf(fmaf(t, w_per[lane - 1], b_per[lane - 1]));
  float acc = Bias[lane];
  #pragma unroll 4
  for (int j = 0; j < NET; ++j)
    acc = fmaf(__shfl(te, j, 32), W[j * NET + lane], acc);
  out[(size_t)row * NET + lane] = acc;
}

// =====================================================================
// Kernel 3: attention. One block per (b,h); 8 waves, one 16-row q-strip
// each.  Scores via WMMA K=4 (ETK), softmax-exp in LDS, then three
// WMMA accumulation chains over LK=256 (64 K=4 steps):
//   NUM1 = E @ MV, Z = E @ Mf, NUM2 = (E*F) @ MD,  x = (NUM1+NUM2)/Z
// =====================================================================
#define ESTRIDE 260   // 256 + 4 pad -> conflict-free A-operand reads
__global__ __launch_bounds__(256) void attn_kernel(
    const float* __restrict__ query_tt, const float* __restrict__ key_tt,
    const float* __restrict__ Qemb, const float* __restrict__ Kemb,
    const float* __restrict__ MV, const float* __restrict__ Mf,
    const float* __restrict__ MD, const float* __restrict__ w_decay,
    float* __restrict__ X)   // [B][LQ][H][D]
{
  __shared__ float sE[8 * 16 * ESTRIDE];   // 130 KB: per-wave 16x256 strips
  const int b = blockIdx.x / NH_, h = blockIdx.x % NH_;
  const int wave = threadIdx.x >> 5, lane = threadIdx.x & 31;
  const int lx = lane & 15, hi = lane >> 4;
  float* strip = sE + wave * 16 * ESTRIDE;
  const int q0 = wave * 16;

  // ---- scores: S(16x256) = Qtile(16x4) x Kemb^T(4x256), scale 1/sqrt(4) ----
  v2f A;
  {
    const float4 qv = *(const float4*)(Qemb + ((size_t)(b * NLQ + q0 + lx) * NET + h * 4));
    A.x = hi ? qv.z : qv.x;
    A.y = hi ? qv.w : qv.y;
  }
  for (int kt = 0; kt < 16; ++kt) {
    const float4 kv = *(const float4*)(Kemb + ((size_t)(b * NLK + kt * 16 + lx) * NET + h * 4));
    v2f Bv; Bv.x = hi ? kv.z : kv.x; Bv.y = hi ? kv.w : kv.y;
    v8f c = {};
    c = wmma_f32(A, Bv, c);
    #pragma unroll
    for (int j = 0; j < 8; ++j)   // D layout: M = j + 8*hi, N = kt*16 + lx
      strip[(j + 8 * hi) * ESTRIDE + kt * 16 + lx] = c[j] * 0.5f;
  }
  __syncthreads();

  // ---- per-row max + exp (row = lx, column half = hi) ----
  {
    float* rowp = strip + lx * ESTRIDE + hi * 128;
    float m = -__builtin_inff();
    for (int c = 0; c < 128; ++c) m = fmaxf(m, rowp[c]);
    m = fmaxf(m, __shfl_xor(m, 16, 32));
    for (int c = 0; c < 128; ++c) rowp[c] = expf(rowp[c] - m);
  }
  __syncthreads();

  // ---- aggregation: 3 WMMA chains over K=256 ----
  const float wd = w_decay[0];
  const float tq = query_tt[b * NLQ + q0 + lx];
  for (int dt = 0; dt < 2; ++dt) {
    v8f acc1 = {}, acc2 = {}, accZ = {};
    const int dcol = dt * 16 + lx;
    for (int ks = 0; ks < 64; ++ks) {
      const int r0 = ks * 4 + 2 * hi, r1 = r0 + 1;   // K indices for this lane
      const float e0 = strip[lx * ESTRIDE + r0];
      const float e1 = strip[lx * ESTRIDE + r1];
      const float dt0 = tq - key_tt[b * NLK + r0];
      const float dt1 = tq - key_tt[b * NLK + r1];
      const float f0 = dt0 / fmaf(dt0, wd, 1.0f);    // dt * time_decay
      const float f1 = dt1 / fmaf(dt1, wd, 1.0f);
      v2f AE; AE.x = e0;      AE.y = e1;
      v2f AF; AF.x = e0 * f0; AF.y = e1 * f1;
      const size_t i0 = (size_t)(b * NLK + r0) * ND + dcol;
      const size_t i1 = (size_t)(b * NLK + r1) * ND + dcol;
      v2f Bmv; Bmv.x = MV[i0]; Bmv.y = MV[i1];
      v2f Bm;  Bm.x  = Mf[i0]; Bm.y  = Mf[i1];
      v2f Bmd; Bmd.x = MD[i0]; Bmd.y = MD[i1];
      acc1 = wmma_f32(AE, Bmv, acc1);
      accZ = wmma_f32(AE, Bm,  accZ);
      acc2 = wmma_f32(AF, Bmd, acc2);
    }
    #pragma unroll
    for (int j = 0; j < 8; ++j) {
      const int q = q0 + j + 8 * hi;
      X[((size_t)(b * NLQ + q) * NH_ + h) * ND + dcol] =
          (acc1[j] + acc2[j]) / accZ[j];
    }
  }
}

// =====================================================================
// Kernel 4: out = X(B*LQ x 256) @ wo(256x64) + bo, WMMA chains.
// One wave per (b, q-tile); loops 4 n-tiles x 64 K-steps.
// =====================================================================
__global__ __launch_bounds__(32) void outproj_kernel(
    const float* __restrict__ X, const float* __restrict__ wo,
    const float* __restrict__ bo, float* __restrict__ out)
{
  const int b = blockIdx.x >> 3, qt = blockIdx.x & 7;
  const int lane = threadIdx.x, lx = lane & 15, hi = lane >> 4;
  const int q0 = qt * 16;
  const float* xrow = X + (size_t)(b * NLQ + q0 + lx) * (NH_ * ND);
  for (int nt = 0; nt < 4; ++nt) {
    const int n = nt * 16 + lx;
    v8f acc = {};
    for (int ks = 0; ks < 64; ++ks) {
      const int r0 = ks * 4 + 2 * hi, r1 = r0 + 1;
      v2f Av; Av.x = xrow[r0];         Av.y = xrow[r1];
      v2f Bv; Bv.x = wo[r0 * NNH + n]; Bv.y = wo[r1 * NNH + n];
      acc = wmma_f32(Av, Bv, acc);
    }
    const float bn = bo[n];
    #pragma unroll
    for (int j = 0; j < 8; ++j)
      out[(size_t)(b * NLQ + q0 + j + 8 * hi) * NNH + n] = acc[j] + bn;
  }
}

// =====================================================================
extern "C" void kernel_launch(void* const* d_in, const int* in_sizes, int n_in,
                              void* d_out, int out_size, void* d_ws, size_t ws_size,
                              hipStream_t stream)
{
  (void)in_sizes; (void)n_in; (void)out_size; (void)ws_size;
  const float* query_tt = (const float*)d_in[0];
  const float* key_tt   = (const float*)d_in[1];
  const float* value    = (const float*)d_in[2];
  const int*   emb_mask = (const int*)  d_in[3];
  const float* w_time   = (const float*)d_in[4];
  const float* b_time   = (const float*)d_in[5];
  const float* w_per    = (const float*)d_in[6];
  const float* b_per    = (const float*)d_in[7];
  const float* w_decay  = (const float*)d_in[8];
  const float* vt_w1    = (const float*)d_in[9];
  const float* vt_b1    = (const float*)d_in[10];
  const float* ln_g     = (const float*)d_in[11];
  const float* ln_b     = (const float*)d_in[12];
  const float* vt_w2    = (const float*)d_in[13];
  const float* vt_b2    = (const float*)d_in[14];
  const float* wq       = (const float*)d_in[15];
  const float* bq       = (const float*)d_in[16];
  const float* wk       = (const float*)d_in[17];
  const float* bk       = (const float*)d_in[18];
  const float* wo       = (const float*)d_in[19];
  const float* bo       = (const float*)d_in[20];

  float* ws   = (float*)d_ws;
  float* Qemb = ws; ws += NB * NLQ * NET;       // 32768
  float* Kemb = ws; ws += NB * NLK * NET;       // 65536
  float* MV   = ws; ws += NB * NLK * ND;        // 65536
  float* Mfv  = ws; ws += NB * NLK * ND;        // 65536
  float* MD   = ws; ws += NB * NLK * ND;        // 65536
  float* X    = ws; ws += NB * NLQ * NH_ * ND;  // 262144  (~2.2 MB total)

  prep_value_kernel<<<NB * NLK, 32, 0, stream>>>(
      value, emb_mask, vt_w1, vt_b1, ln_g, ln_b, vt_w2, vt_b2, MV, Mfv, MD);
  time_emb_kernel<<<NB * NLQ, 32, 0, stream>>>(
      query_tt, w_time, b_time, w_per, b_per, wq, bq, Qemb);
  time_emb_kernel<<<NB * NLK, 32, 0, stream>>>(
      key_tt, w_time, b_time, w_per, b_per, wk, bk, Kemb);
  attn_kernel<<<NB * NH_, 256, 0, stream>>>(
      query_tt, key_tt, Qemb, Kemb, MV, Mfv, MD, w_decay, X);
  outproj_kernel<<<NB * 8, 32, 0, stream>>>(X, wo, bo, (float*)d_out);
}